// DeChunkLayer_292057776376
// MI455X (gfx1250) — compile-verified
//
#include <hip/hip_runtime.h>
#include <hip/hip_bf16.h>

// Problem constants (from reference): B=4, L=8192, D=1024
#define BB 4
#define LL 8192
#define DD 1024
#define TT 16            // WMMA chunk length (M = K = 16)
#define CC (LL / TT)     // 512 chunks per batch
#define EPSV 1e-4f

typedef __attribute__((ext_vector_type(2))) float v2f;
typedef __attribute__((ext_vector_type(8))) float v8f;

// ---------------------------------------------------------------------------
// Kernel 1: idx[b,t] = cumsum(mask)-1  (block-per-batch scan, 256 thr x 32 el)
// ---------------------------------------------------------------------------
__global__ __launch_bounds__(256) void k_idx(const unsigned char* __restrict__ mask,
                                             int* __restrict__ idx) {
  __shared__ int part[256];
  __shared__ int excl[256];
  const int b = blockIdx.x;
  const int tid = threadIdx.x;
  const unsigned char* m = mask + (size_t)b * LL;
  const int base = tid * 32;
  int loc[32];
  int s = 0;
#pragma unroll
  for (int i = 0; i < 32; ++i) {
    loc[i] = m[base + i] ? 1 : 0;
    s += loc[i];
  }
  part[tid] = s;
  __syncthreads();
  if (tid == 0) {
    int run = 0;
    for (int j = 0; j < 256; ++j) { excl[j] = run; run += part[j]; }
  }
  __syncthreads();
  int run = excl[tid];
  int* op = idx + (size_t)b * LL + base;
#pragma unroll
  for (int i = 0; i < 32; ++i) {
    run += loc[i];
    op[i] = run - 1;
  }
}

// ---------------------------------------------------------------------------
// Kernel 2: per 16-chunk coefficients.
//   Wp[t][s] = p_s * prod_{i=s+1..t} a_i   (s<=t, else 0)   [p folded in]
//   Pc[t]    = prod_{i=0..t} a_i           (carry weights; A_c = Pc[15])
// One thread per chunk (B*C = 2048 threads). No divisions (underflow-safe).
// ---------------------------------------------------------------------------
__global__ __launch_bounds__(256) void k_coef(const float* __restrict__ prob,
                                              float* __restrict__ Wp,
                                              float* __restrict__ Pc) {
  const int g = blockIdx.x * blockDim.x + threadIdx.x;   // b*CC + c
  if (g >= BB * CC) return;
  const int b = g >> 9;          // /CC
  const int c = g & (CC - 1);
  const float* pp = prob + (((size_t)b * LL + (size_t)c * TT) * 2) + 1;  // [...,1]
  float p[TT], a[TT];
#pragma unroll
  for (int s = 0; s < TT; ++s) {
    float v = pp[2 * s];
    v = fminf(fmaxf(v, EPSV), 1.0f - EPSV);
    if (c == 0 && s == 0) v = 1.0f;          // first token forced boundary
    p[s] = v;
    a[s] = 1.0f - v;
  }
  float* w = Wp + (size_t)g * (TT * TT);
  float* pc = Pc + (size_t)g * TT;
  float run = 1.0f;
#pragma unroll
  for (int t = 0; t < TT; ++t) { run *= a[t]; pc[t] = run; }
#pragma unroll
  for (int s = 0; s < TT; ++s) {
    float ww = p[s];
    w[s * TT + s] = ww;
#pragma unroll
    for (int t = s + 1; t < TT; ++t) { ww *= a[t]; w[t * TT + s] = ww; }
#pragma unroll
    for (int t = 0; t < s; ++t) w[t * TT + s] = 0.0f;
  }
}

// ---------------------------------------------------------------------------
// Kernel 3: zero-carry chunk end state: end[b,c,d] = sum_s Wp[15][s] * x~[s,d]
// grid = (B*C, D/256), 256 threads; coalesced over d.
// ---------------------------------------------------------------------------
__global__ __launch_bounds__(256) void k_ends(const float* __restrict__ x,
                                              const int* __restrict__ idx,
                                              const float* __restrict__ Wp,
                                              float* __restrict__ ends) {
  const int bx = blockIdx.x;                 // b*CC + c
  const int b = bx >> 9;
  const int c = bx & (CC - 1);
  const int d = blockIdx.y * 256 + threadIdx.x;
  const float* w15 = Wp + (size_t)bx * (TT * TT) + 15 * TT;
  const int* ip = idx + (size_t)b * LL + (size_t)c * TT;
  const float* xb = x + (size_t)b * LL * DD;
  float acc = 0.0f;
#pragma unroll
  for (int s = 0; s < TT; ++s) {
    acc = fmaf(w15[s], xb[(size_t)ip[s] * DD + d], acc);
  }
  ends[(size_t)bx * DD + d] = acc;
}

// ---------------------------------------------------------------------------
// Kernel 4: sequential inter-chunk carry scan (in place):
//   read e = end[c][d]; write end[c][d] = carry_in; carry = A_c*carry + e
// One thread per (b,d); unrolled x8 for MLP. grid=(D/256, B), 256 threads.
// ---------------------------------------------------------------------------
__global__ __launch_bounds__(256) void k_scan(float* __restrict__ ends,
                                              const float* __restrict__ Pc) {
  const int d = blockIdx.x * 256 + threadIdx.x;
  const int b = blockIdx.y;
  float* e = ends + (size_t)b * CC * DD + d;
  const float* A = Pc + (size_t)b * CC * TT + 15;   // Pc[c][15] = A_c
  float cur = 0.0f;
  for (int c0 = 0; c0 < CC; c0 += 8) {
    float ev[8], Av[8];
#pragma unroll
    for (int i = 0; i < 8; ++i) Av[i] = A[(size_t)(c0 + i) * TT];
#pragma unroll
    for (int i = 0; i < 8; ++i) ev[i] = e[(size_t)(c0 + i) * DD];
#pragma unroll
    for (int i = 0; i < 8; ++i) {
      e[(size_t)(c0 + i) * DD] = cur;               // carry entering chunk c0+i
      cur = fmaf(Av[i], cur, ev[i]);
    }
  }
}

// ---------------------------------------------------------------------------
// Kernel 5: main WMMA pass. One wave per (b, chunk, 16-wide N tile).
//   out[16x16 tile] = Wp(16x16) @ x~(16x16) + Pc[t] (outer) carry[n]
// via 4 chained V_WMMA_F32_16X16X4_F32, accumulator seeded with carry term.
// Layouts (per CDNA5 ISA 7.12.2, f32 shapes):
//   A 16x4 : lane r=l&15 row; VGPR j holds K = j + 2*(l>>4)
//   B 4x16 : lane n=l&15 col; VGPR j holds K = j + 2*(l>>4)
//   C 16x16: VGPR j, lanes0-15 -> row j; lanes16-31 -> row j+8
// ---------------------------------------------------------------------------
__global__ __launch_bounds__(256) void k_wmma(const float* __restrict__ x,
                                              const int* __restrict__ idx,
                                              const float* __restrict__ Wp,
                                              const float* __restrict__ Pc,
                                              const float* __restrict__ carry,
                                              float* __restrict__ out) {
  const int gtid = blockIdx.x * 256 + threadIdx.x;
  const int wid  = gtid >> 5;                 // global wave id, [0, B*CC*64)
  const int lane = threadIdx.x & 31;
  const int n0 = (wid & 63) << 4;             // N tile base (64 tiles of 16)
  const int c  = (wid >> 6) & (CC - 1);
  const int b  = wid >> 15;
  const int h = lane >> 4;                    // lane half
  const int r = lane & 15;                    // A row / B,C column index
  const int bc = b * CC + c;

  const float* wbase = Wp + (size_t)bc * (TT * TT);
  const float* pcb   = Pc + (size_t)bc * TT + h * 8;     // rows h*8 .. h*8+7
  const int*   ip    = idx + (size_t)b * LL + (size_t)c * TT;
  const float* xb    = x + (size_t)b * LL * DD + n0 + r;
  const float  vc    = carry[(size_t)bc * DD + n0 + r];  // carry entering chunk

  v8f acc;
#pragma unroll
  for (int j = 0; j < 8; ++j) acc[j] = pcb[j] * vc;      // seed: Pc[t]*carry[n]

#pragma unroll
  for (int q = 0; q < 4; ++q) {
    const int k0 = 4 * q + 2 * h;
    v2f afr = { wbase[r * TT + k0], wbase[r * TT + k0 + 1] };
    const int i0 = ip[k0];
    const int i1 = ip[k0 + 1];
    v2f bfr = { xb[(size_t)i0 * DD], xb[(size_t)i1 * DD] };
    acc = __builtin_amdgcn_wmma_f32_16x16x4_f32(
        /*neg_a=*/false, afr, /*neg_b=*/false, bfr,
        /*c_mod=*/(short)0, acc, /*reuse_a=*/false, /*reuse_b=*/false);
  }

  float* ob = out + ((size_t)b * LL + (size_t)c * TT) * DD + n0 + r;
#pragma unroll
  for (int j = 0; j < 8; ++j) ob[(size_t)(j + 8 * h) * DD] = acc[j];
}

// ---------------------------------------------------------------------------
extern "C" void kernel_launch(void* const* d_in, const int* in_sizes, int n_in,
                              void* d_out, int out_size, void* d_ws, size_t ws_size,
                              hipStream_t stream) {
  const float*         x    = (const float*)d_in[0];          // (B,L,D) f32
  const unsigned char* mask = (const unsigned char*)d_in[1];  // (B,L) bool (1B)
  const float*         prob = (const float*)d_in[2];          // (B,L,2) f32
  float* out = (float*)d_out;

  // Workspace layout
  char* ws = (char*)d_ws;
  int*   idx  = (int*)ws;                                   // B*L*4      = 128 KB
  float* Wp   = (float*)(ws + (size_t)BB * LL * 4);         // B*C*256*4  = 2 MB
  float* Pc   = (float*)(ws + (size_t)BB * LL * 4
                            + (size_t)BB * CC * TT * TT * 4);   // B*C*16*4 = 128 KB
  float* ends = (float*)(ws + (size_t)BB * LL * 4
                            + (size_t)BB * CC * TT * TT * 4
                            + (size_t)BB * CC * TT * 4);        // B*C*D*4  = 8 MB

  k_idx <<<BB, 256, 0, stream>>>(mask, idx);
  k_coef<<<(BB * CC) / 256, 256, 0, stream>>>(prob, Wp, Pc);
  k_ends<<<dim3(BB * CC, DD / 256), 256, 0, stream>>>(x, idx, Wp, ends);
  k_scan<<<dim3(DD / 256, BB), 256, 0, stream>>>(ends, Pc);
  // 'ends' now holds per-chunk carry-in. 131072 waves -> 16384 blocks of 8 waves.
  k_wmma<<<(BB * CC * 64 * 32) / 256, 256, 0, stream>>>(x, idx, Wp, Pc, ends, out);
}